// KronLinear_80006650790412
// MI455X (gfx1250) — compile-verified
//
#include <hip/hip_runtime.h>

typedef __attribute__((ext_vector_type(2))) float v2f;
typedef __attribute__((ext_vector_type(8))) float v8f;

#define RANK 16
#define DIM  64
#define LDSP 68   // padded LDS row stride (floats): multiple of 4 (float4 stores), !=64 (bank conflicts)

// ---------------------------------------------------------------------------
// Prep: build scrambled B' and transposed A into workspace (L2-resident after).
//   Bp[rv][j][b2]  = b[(rv*64+b2)%16][j][(rv*64+b2)/16]   (the reference's re-view)
//   At[rv][a2][i]  = a[rv][i][a2]
// ---------------------------------------------------------------------------
__global__ __launch_bounds__(256) void kron_prep(const float* __restrict__ a,
                                                 const float* __restrict__ b,
                                                 float* __restrict__ Bp,
                                                 float* __restrict__ At)
{
    int idx = blockIdx.x * 256 + threadIdx.x;          // 0 .. 16*4096-1
    if (idx >= RANK * DIM * DIM) return;
    int rv  = idx >> 12;
    int rem = idx & 4095;
    int j   = rem >> 6;        // dim0 of the 64x64 slice
    int c   = rem & 63;        // dim1 of the 64x64 slice
    int k   = rv * 64 + c;
    Bp[idx] = b[(k & 15) * 4096 + j * 64 + (k >> 4)];
    At[idx] = a[rv * 4096 + c * 64 + j];               // transpose within slice
}

// ---------------------------------------------------------------------------
// Main: one 256-thread (8-wave) workgroup per 64x64 tile of x.
//   O[n] = sum_r At_r * (X_n * Bp_r) + bias,  all fp32 via v_wmma_f32_16x16x4_f32
// ---------------------------------------------------------------------------
__global__ __launch_bounds__(256) void kron_wmma(const float* __restrict__ x,
                                                 const float* __restrict__ Bp,
                                                 const float* __restrict__ At,
                                                 const float* __restrict__ bias,
                                                 float* __restrict__ out)
{
    extern __shared__ float smem[];
    float* sX = smem;                  // 64 x LDSP : X tile (i,j)
    float* sT = sX + DIM * LDSP;       // 64 x LDSP : T = X*Bp_r (i,b2)
    float* sB = sT + DIM * LDSP;       // 64 x LDSP : Bp_r (j,b2)
    float* sA = sB + DIM * LDSP;       // 64 x LDSP : At_r (a2,i)

    const int tid  = threadIdx.x;
    const int lane = tid & 31;
    const int wave = tid >> 5;         // 0..7
    const int rlo  = lane & 15;        // row/col-within-tile index
    const int hi   = lane >> 4;        // half-wave select (K pair / M+8)
    const int nt   = wave & 3;         // output column tile 0..3
    const int mt0  = wave >> 2;        // output row tiles mt0 and mt0+2

    const size_t base = (size_t)blockIdx.x * 4096;
    const float* xblk = x + base;

    // cooperative load of X (1024 float4 / 256 threads)
    #pragma unroll
    for (int i = 0; i < 4; ++i) {
        int e4  = tid + i * 256;
        int e   = e4 * 4;
        int row = e >> 6, col = e & 63;
        *(float4*)&sX[row * LDSP + col] = ((const float4*)xblk)[e4];
    }

    v8f oacc[2] = { v8f{}, v8f{} };

    for (int r = 0; r < RANK; ++r) {
        __syncthreads();   // previous iteration finished reading sB/sA/sT
        {
            const float* bsrc = Bp + r * 4096;
            const float* asrc = At + r * 4096;
            #pragma unroll
            for (int i = 0; i < 4; ++i) {
                int e4  = tid + i * 256;
                int e   = e4 * 4;
                int row = e >> 6, col = e & 63;
                *(float4*)&sB[row * LDSP + col] = ((const float4*)bsrc)[e4];
                *(float4*)&sA[row * LDSP + col] = ((const float4*)asrc)[e4];
            }
        }
        __syncthreads();

        // ---- stage 1: T = X * Bp_r  (each wave: 2 of the 16 output tiles) ----
        #pragma unroll
        for (int half = 0; half < 2; ++half) {
            const int mt = mt0 + half * 2;
            v8f t = {};
            #pragma unroll
            for (int kk = 0; kk < 16; ++kk) {
                const int kb = kk * 4 + hi * 2;   // this half-wave's K pair
                v2f af, bf;
                af.x = sX[(mt * 16 + rlo) * LDSP + kb];
                af.y = sX[(mt * 16 + rlo) * LDSP + kb + 1];
                bf.x = sB[kb * LDSP + nt * 16 + rlo];
                bf.y = sB[(kb + 1) * LDSP + nt * 16 + rlo];
                t = __builtin_amdgcn_wmma_f32_16x16x4_f32(
                        false, af, false, bf, (short)0, t, false, false);
            }
            // spill T tile to LDS per C/D layout: VGPR v -> row v + 8*hi
            #pragma unroll
            for (int v = 0; v < 8; ++v)
                sT[(mt * 16 + v + 8 * hi) * LDSP + nt * 16 + rlo] = t[v];
        }
        __syncthreads();

        // ---- stage 2: O += At_r * T ----
        #pragma unroll
        for (int half = 0; half < 2; ++half) {
            const int mt = mt0 + half * 2;
            v8f o = oacc[half];
            #pragma unroll
            for (int kk = 0; kk < 16; ++kk) {
                const int kb = kk * 4 + hi * 2;
                v2f af, bf;
                af.x = sA[(mt * 16 + rlo) * LDSP + kb];
                af.y = sA[(mt * 16 + rlo) * LDSP + kb + 1];
                bf.x = sT[kb * LDSP + nt * 16 + rlo];
                bf.y = sT[(kb + 1) * LDSP + nt * 16 + rlo];
                o = __builtin_amdgcn_wmma_f32_16x16x4_f32(
                        false, af, false, bf, (short)0, o, false, false);
            }
            oacc[half] = o;
        }
    }

    // ---- epilogue: out[n][a2][b2] = O[a2][b2] + bias[a2*64+b2] ----
    float* oblk = out + base;
    #pragma unroll
    for (int half = 0; half < 2; ++half) {
        const int mt = mt0 + half * 2;
        #pragma unroll
        for (int v = 0; v < 8; ++v) {
            const int a2 = mt * 16 + v + 8 * hi;
            const int b2 = nt * 16 + rlo;
            oblk[a2 * 64 + b2] = oacc[half][v] + bias[a2 * 64 + b2];
        }
    }
}

// ---------------------------------------------------------------------------
extern "C" void kernel_launch(void* const* d_in, const int* in_sizes, int n_in,
                              void* d_out, int out_size, void* d_ws, size_t ws_size,
                              hipStream_t stream)
{
    const float* x    = (const float*)d_in[0];
    const float* a    = (const float*)d_in[1];
    const float* b    = (const float*)d_in[2];
    const float* bias = (const float*)d_in[3];
    float* out = (float*)d_out;

    float* Bp = (float*)d_ws;                 // 16*4096 floats
    float* At = Bp + RANK * DIM * DIM;        // 16*4096 floats (512 KB total)

    kron_prep<<<(RANK * DIM * DIM + 255) / 256, 256, 0, stream>>>(a, b, Bp, At);

    const int nTiles = out_size / (DIM * DIM);            // 8192
    const size_t smem = (size_t)4 * DIM * LDSP * sizeof(float);  // ~68 KB
    kron_wmma<<<nTiles, 256, smem, stream>>>(x, Bp, At, bias, out);
}